// NullBeamformor_89653147336806
// MI455X (gfx1250) — compile-verified
//
#include <hip/hip_runtime.h>
#include <hip/hip_bf16.h>

#define NUM_BEAM    24
#define NUM_NULL    4
#define NUM_BIN     257
#define NUM_CHANNEL 8
#define N_FRAMES    4096
#define MAX_TILES   280   // sum_b ceil(count_b/16) <= 4096/16 + 24 = 280

// Workspace layout (int32 slots)
#define WS_NUMTILES 0
#define WS_COUNT    16            // 24 counts
#define WS_CURSOR   48            // 24 cursors
#define WS_OFFSET   80            // 24 offsets
#define WS_TSTART   112           // MAX_TILES
#define WS_TLEN     (112 + MAX_TILES)          // MAX_TILES
#define WS_FLIST    (112 + 2 * MAX_TILES)      // N_FRAMES
// total ints = 112 + 2*280 + 4096 = 4768  (~19 KB)

typedef __attribute__((ext_vector_type(2))) float v2f;
typedef __attribute__((ext_vector_type(8))) float v8f;

__global__ __launch_bounds__(32)
void nb_init(int* ws) {
    int i = threadIdx.x;
    if (i < NUM_BEAM) { ws[WS_COUNT + i] = 0; ws[WS_CURSOR + i] = 0; }
    if (i == 0) ws[WS_NUMTILES] = 0;
}

__global__ __launch_bounds__(256)
void nb_count(const int* __restrict__ beam_id, int* ws) {
    int n = blockIdx.x * blockDim.x + threadIdx.x;
    if (n < N_FRAMES) atomicAdd(&ws[WS_COUNT + beam_id[n]], 1);
}

__global__ __launch_bounds__(32)
void nb_scan(int* ws) {
    if (threadIdx.x != 0) return;
    int off = 0, t = 0;
    for (int b = 0; b < NUM_BEAM; ++b) {
        ws[WS_OFFSET + b] = off;
        int c = ws[WS_COUNT + b];
        for (int s = 0; s < c; s += 16) {
            ws[WS_TSTART + t] = off + s;
            ws[WS_TLEN + t]   = (c - s < 16) ? (c - s) : 16;
            ++t;
        }
        off += c;
    }
    ws[WS_NUMTILES] = t;
}

__global__ __launch_bounds__(256)
void nb_scatter(const int* __restrict__ beam_id, int* ws) {
    int n = blockIdx.x * blockDim.x + threadIdx.x;
    if (n >= N_FRAMES) return;
    int b = beam_id[n];
    int pos = ws[WS_OFFSET + b] + atomicAdd(&ws[WS_CURSOR + b], 1);
    ws[WS_FLIST + pos] = n;
}

// One wave (32 lanes) per (frame-tile of 16, frequency f).
// Complex beamforming folded into a real GEMM:
//   A (16x16): row m = frame, K' = [xr(0..7), xi(0..7)]
//   B (16x8) : col j<4 -> real_j = [wr_j ; -wi_j], j in [4,8) -> imag = [wi ; wr]
// Contraction K=16 done as 4 chained v_wmma_f32_16x16x4_f32 (exact fp32).
__global__ __launch_bounds__(32)
void nb_main(const float* __restrict__ x, const float* __restrict__ W,
             const int* __restrict__ beam_id, float* __restrict__ out,
             const int* __restrict__ ws) {
    int t = blockIdx.x;
    if (t >= ws[WS_NUMTILES]) return;
    int f     = blockIdx.y;
    int start = ws[WS_TSTART + t];
    int len   = ws[WS_TLEN + t];

    int l    = threadIdx.x;
    int half = l >> 4;     // lane group: K offset +2
    int mrow = l & 15;     // A row (frame within tile)
    int ncol = l & 15;     // B/D column

    const int* flist = ws + WS_FLIST;
    int b  = beam_id[flist[start]];               // all frames in tile share beam
    int nm = flist[start + (mrow < len ? mrow : len - 1)];

    // ---- A: lane holds A[mrow, kb], A[mrow, kb+1], kb = 4g + 2*half ----
    // kb<8 -> xr channel kb ; kb>=8 -> xi channel kb-8 (pairs never cross).
    const float* xbase = x + ((size_t)nm * 2 * NUM_BIN + f) * NUM_CHANNEL;
    v2f a[4];
#pragma unroll
    for (int g = 0; g < 4; ++g) {
        int kb = 4 * g + 2 * half;            // even
        int ri = kb >> 3;
        int c  = kb & 7;
        a[g] = *(const v2f*)(xbase + (size_t)ri * NUM_BIN * NUM_CHANNEL + c);
    }

    // ---- B: unconditional loads; invalid columns (ncol>=8) scaled to 0 ----
    int   nullk = ncol & 3;
    int   riout = (ncol >> 2) & 1;            // 0 = real column, 1 = imag column
    float cmask = (ncol < 8) ? 1.0f : 0.0f;   // zero out unused D columns
    const float* wbase = W + (((size_t)(b * NUM_NULL + nullk) * 2) * NUM_BIN + f) * NUM_CHANNEL;
    v2f bm[4];
#pragma unroll
    for (int g = 0; g < 4; ++g) {
        int   kb  = 4 * g + 2 * half;
        int   hi  = kb >> 3;                  // 0: xr partial, 1: xi partial
        int   c   = kb & 7;
        int   riw = hi ? (1 - riout) : riout; // which W component to read
        float sgn = (hi && riout == 0) ? -1.0f : 1.0f;  // real col xi part: -wi
        v2f w = *(const v2f*)(wbase + (size_t)riw * NUM_BIN * NUM_CHANNEL + c);
        bm[g] = w * (sgn * cmask);
    }

    // ---- 4 chained fp32 WMMAs ----
    v8f acc = {};
#pragma unroll
    for (int g = 0; g < 4; ++g) {
        acc = __builtin_amdgcn_wmma_f32_16x16x4_f32(
            /*neg_a=*/false, a[g], /*neg_b=*/false, bm[g],
            /*c_mod=*/(short)0, acc, /*reuse_a=*/false, /*reuse_b=*/false);
    }

    // ---- D -> LDS (16x16, cols 8..15 are scratch) -> float4 stores ----
    __shared__ float tile[16][16];
#pragma unroll
    for (int d = 0; d < 8; ++d)
        tile[d + 8 * half][ncol] = acc[d];    // D: M = d + 8*(lane/16), N = lane%16
    __syncthreads();

    int m2 = l >> 1, ri = l & 1;              // lane -> (frame, real/imag half)
    if (m2 < len) {
        int n = flist[start + m2];
        float4 v = *(const float4*)&tile[m2][ri * 4];
        *(float4*)(out + (((size_t)n * 2 + ri) * NUM_BIN + f) * NUM_NULL) = v;
    }
}

extern "C" void kernel_launch(void* const* d_in, const int* in_sizes, int n_in,
                              void* d_out, int out_size, void* d_ws, size_t ws_size,
                              hipStream_t stream) {
    (void)in_sizes; (void)n_in; (void)out_size; (void)ws_size;
    const float* x       = (const float*)d_in[0];
    const float* W       = (const float*)d_in[1];
    const int*   beam_id = (const int*)d_in[2];
    float*       out     = (float*)d_out;
    int*         ws      = (int*)d_ws;

    nb_init<<<1, 32, 0, stream>>>(ws);
    nb_count<<<(N_FRAMES + 255) / 256, 256, 0, stream>>>(beam_id, ws);
    nb_scan<<<1, 32, 0, stream>>>(ws);
    nb_scatter<<<(N_FRAMES + 255) / 256, 256, 0, stream>>>(beam_id, ws);

    dim3 grid(MAX_TILES, NUM_BIN, 1);
    nb_main<<<grid, 32, 0, stream>>>(x, W, beam_id, out, ws);
}